// GAttConv_88828513616037
// MI455X (gfx1250) — compile-verified
//
#include <hip/hip_runtime.h>
#include <math.h>

typedef __attribute__((ext_vector_type(2))) float v2f;
typedef __attribute__((ext_vector_type(8))) float v8f;

#define F_DIM 128
#define LDSP  (F_DIM + 4)      // padded LDS row stride (132 -> conflict-free)

// ---------------------------------------------------------------------------
// K1: proj = feat @ w1_half   (blockIdx.y = 0 -> w1 rows [0,128) = proj_dst,
//                              blockIdx.y = 1 -> w1 rows [128,256) = proj_src)
// 64 output rows per block (4 x 16-row WMMA tiles). 8 waves = 128 cols.
// B panel (128x16 per wave) hoisted into registers once, reused by 4 tiles.
// ---------------------------------------------------------------------------
__global__ __launch_bounds__(256) void proj_kernel(const float* __restrict__ feat,
                                                   const float* __restrict__ w1,
                                                   float* __restrict__ proj,
                                                   int n_nodes) {
    __shared__ __align__(16) float As[64][LDSP];   // 64*132*4 = 33792 B

    const int row0  = blockIdx.x * 64;
    const int which = blockIdx.y;
    const float* W  = w1 + (size_t)which * F_DIM * F_DIM;
    float* out      = proj + (size_t)which * (size_t)n_nodes * F_DIM;

    const int t = threadIdx.x;
    // Stage A tile (64 rows x 128 f32) into LDS, float4-coalesced.
    for (int i = t; i < 64 * (F_DIM / 4); i += 256) {
        int row = i >> 5;          // 32 float4 per row
        int c4  = i & 31;
        float4 v = make_float4(0.f, 0.f, 0.f, 0.f);
        if (row0 + row < n_nodes)
            v = ((const float4*)(feat + (size_t)(row0 + row) * F_DIM))[c4];
        *(float4*)&As[row][c4 * 4] = v;
    }
    __syncthreads();

    const int wave = t >> 5;
    const int lane = t & 31;
    const int half = lane >> 4;
    const int r    = lane & 15;
    const int col0 = wave * 16;

    // Hoist B panel into registers (WMMA 16x16x4 B layout: lane half picks K pair)
    float breg[64];
#pragma unroll
    for (int k0 = 0; k0 < F_DIM; k0 += 4) {
        const int ka = k0 + 2 * half;
        breg[(k0 >> 2) * 2 + 0] = W[(size_t)ka * F_DIM + col0 + r];
        breg[(k0 >> 2) * 2 + 1] = W[(size_t)(ka + 1) * F_DIM + col0 + r];
    }

    v8f acc[4];
#pragma unroll
    for (int tile = 0; tile < 4; ++tile)
        acc[tile] = (v8f){0.f, 0.f, 0.f, 0.f, 0.f, 0.f, 0.f, 0.f};

#pragma unroll
    for (int k0 = 0; k0 < F_DIM; k0 += 4) {
        const int ka = k0 + 2 * half;
        v2f b;
        b.x = breg[(k0 >> 2) * 2 + 0];
        b.y = breg[(k0 >> 2) * 2 + 1];
#pragma unroll
        for (int tile = 0; tile < 4; ++tile) {
            v2f a;
            a.x = As[tile * 16 + r][ka];
            a.y = As[tile * 16 + r][ka + 1];
            acc[tile] = __builtin_amdgcn_wmma_f32_16x16x4_f32(false, a, false, b,
                                                              (short)0, acc[tile],
                                                              false, false);
        }
    }

#pragma unroll
    for (int tile = 0; tile < 4; ++tile) {
#pragma unroll
        for (int j = 0; j < 8; ++j) {
            int row = row0 + tile * 16 + j + 8 * half;
            if (row < n_nodes)
                out[(size_t)row * F_DIM + col0 + r] = acc[tile][j];
        }
    }
}

// ---------------------------------------------------------------------------
// K2: per-edge attention score. One wave32 per edge; each lane owns 4 features.
// scores[e] = dot(tanh(proj_dst[dst] + proj_src[src] + b1), w2) + b2
// ---------------------------------------------------------------------------
__global__ __launch_bounds__(256) void score_kernel(const float* __restrict__ proj_dst,
                                                    const float* __restrict__ proj_src,
                                                    const float* __restrict__ b1,
                                                    const float* __restrict__ w2,
                                                    const float* __restrict__ b2,
                                                    const int* __restrict__ src,
                                                    const int* __restrict__ dst,
                                                    float* __restrict__ scores,
                                                    int n_edges) {
    const int e    = (int)((blockIdx.x * (size_t)blockDim.x + threadIdx.x) >> 5);
    const int lane = threadIdx.x & 31;
    if (e >= n_edges) return;

    const int s = src[e];
    const int d = dst[e];
    float4 pd = ((const float4*)(proj_dst + (size_t)d * F_DIM))[lane];
    float4 ps = ((const float4*)(proj_src + (size_t)s * F_DIM))[lane];
    float4 bb = ((const float4*)b1)[lane];
    float4 ww = ((const float4*)w2)[lane];

    float sum = tanhf(pd.x + ps.x + bb.x) * ww.x
              + tanhf(pd.y + ps.y + bb.y) * ww.y
              + tanhf(pd.z + ps.z + bb.z) * ww.z
              + tanhf(pd.w + ps.w + bb.w) * ww.w;
#pragma unroll
    for (int off = 16; off > 0; off >>= 1)
        sum += __shfl_xor(sum, off, 32);

    if (lane == 0) scores[e] = sum + b2[0];
}

// ---------------------------------------------------------------------------
// K3: row_ptr[n] = lower_bound(dst, n)  (dst is sorted). row_ptr[N] = E.
// ---------------------------------------------------------------------------
__global__ __launch_bounds__(256) void rowptr_kernel(const int* __restrict__ dst,
                                                     int* __restrict__ row_ptr,
                                                     int n_nodes, int n_edges) {
    const int n = blockIdx.x * blockDim.x + threadIdx.x;
    if (n > n_nodes) return;
    int lo = 0, hi = n_edges;
    while (lo < hi) {
        int mid = (lo + hi) >> 1;
        if (dst[mid] < n) lo = mid + 1; else hi = mid;
    }
    row_ptr[n] = lo;
}

// ---------------------------------------------------------------------------
// K4: segment softmax + weighted aggregation. One wave32 per node.
// neigh[n,:] = sum_e softmax(scores)_e * feat[src[e],:]
// ---------------------------------------------------------------------------
__global__ __launch_bounds__(256) void agg_kernel(const float* __restrict__ feat,
                                                  const float* __restrict__ scores,
                                                  const int* __restrict__ src,
                                                  const int* __restrict__ row_ptr,
                                                  float* __restrict__ neigh,
                                                  int n_nodes) {
    const int node = (int)((blockIdx.x * (size_t)blockDim.x + threadIdx.x) >> 5);
    const int lane = threadIdx.x & 31;
    if (node >= n_nodes) return;

    const int beg = row_ptr[node];
    const int end = row_ptr[node + 1];
    float4 acc = make_float4(0.f, 0.f, 0.f, 0.f);

    if (beg < end) {
        // pass 1: segment max
        float m = -INFINITY;
        for (int e = beg + lane; e < end; e += 32) m = fmaxf(m, scores[e]);
#pragma unroll
        for (int off = 16; off > 0; off >>= 1) m = fmaxf(m, __shfl_xor(m, off, 32));
        // pass 2: denominator
        float den = 0.f;
        for (int e = beg + lane; e < end; e += 32) den += expf(scores[e] - m);
#pragma unroll
        for (int off = 16; off > 0; off >>= 1) den += __shfl_xor(den, off, 32);
        const float inv = 1.0f / den;
        // pass 3: weighted gather (lane owns 4 feature columns -> coalesced float4)
        for (int e = beg; e < end; ++e) {
            const float a = expf(scores[e] - m) * inv;   // lane-uniform
            const int s  = src[e];
            float4 f = ((const float4*)(feat + (size_t)s * F_DIM))[lane];
            acc.x += a * f.x; acc.y += a * f.y; acc.z += a * f.z; acc.w += a * f.w;
        }
    }
    ((float4*)(neigh + (size_t)node * F_DIM))[lane] = acc;
}

// ---------------------------------------------------------------------------
// K5: rst = feat @ wf[:128] + neigh @ wf[128:] + bf
// 32 output rows per block (2 x 16-row WMMA tiles). B register bank (64 f32)
// reused sequentially: Wtop pass over As, then Wbot pass over Bs.
// ---------------------------------------------------------------------------
__global__ __launch_bounds__(256) void out_kernel(const float* __restrict__ feat,
                                                  const float* __restrict__ neigh,
                                                  const float* __restrict__ wf,
                                                  const float* __restrict__ bf,
                                                  float* __restrict__ out,
                                                  int n_nodes) {
    __shared__ __align__(16) float As[32][LDSP];
    __shared__ __align__(16) float Bs[32][LDSP];

    const int row0 = blockIdx.x * 32;
    const int t = threadIdx.x;
    for (int i = t; i < 32 * (F_DIM / 4); i += 256) {
        int row = i >> 5;
        int c4  = i & 31;
        float4 va = make_float4(0.f, 0.f, 0.f, 0.f);
        float4 vb = va;
        if (row0 + row < n_nodes) {
            va = ((const float4*)(feat  + (size_t)(row0 + row) * F_DIM))[c4];
            vb = ((const float4*)(neigh + (size_t)(row0 + row) * F_DIM))[c4];
        }
        *(float4*)&As[row][c4 * 4] = va;
        *(float4*)&Bs[row][c4 * 4] = vb;
    }
    __syncthreads();

    const int wave = t >> 5;
    const int lane = t & 31;
    const int half = lane >> 4;
    const int r    = lane & 15;
    const int col0 = wave * 16;

    const float* Wtop = wf;
    const float* Wbot = wf + (size_t)F_DIM * F_DIM;

    v8f acc[2];
    acc[0] = (v8f){0.f, 0.f, 0.f, 0.f, 0.f, 0.f, 0.f, 0.f};
    acc[1] = (v8f){0.f, 0.f, 0.f, 0.f, 0.f, 0.f, 0.f, 0.f};

    float breg[64];

    // ---- pass 1: feat @ wf_top ----
#pragma unroll
    for (int k0 = 0; k0 < F_DIM; k0 += 4) {
        const int ka = k0 + 2 * half;
        breg[(k0 >> 2) * 2 + 0] = Wtop[(size_t)ka * F_DIM + col0 + r];
        breg[(k0 >> 2) * 2 + 1] = Wtop[(size_t)(ka + 1) * F_DIM + col0 + r];
    }
#pragma unroll
    for (int k0 = 0; k0 < F_DIM; k0 += 4) {
        const int ka = k0 + 2 * half;
        v2f b;
        b.x = breg[(k0 >> 2) * 2 + 0];
        b.y = breg[(k0 >> 2) * 2 + 1];
#pragma unroll
        for (int tile = 0; tile < 2; ++tile) {
            v2f a;
            a.x = As[tile * 16 + r][ka];
            a.y = As[tile * 16 + r][ka + 1];
            acc[tile] = __builtin_amdgcn_wmma_f32_16x16x4_f32(false, a, false, b,
                                                              (short)0, acc[tile],
                                                              false, false);
        }
    }

    // ---- pass 2: neigh @ wf_bot (reuse breg) ----
#pragma unroll
    for (int k0 = 0; k0 < F_DIM; k0 += 4) {
        const int ka = k0 + 2 * half;
        breg[(k0 >> 2) * 2 + 0] = Wbot[(size_t)ka * F_DIM + col0 + r];
        breg[(k0 >> 2) * 2 + 1] = Wbot[(size_t)(ka + 1) * F_DIM + col0 + r];
    }
#pragma unroll
    for (int k0 = 0; k0 < F_DIM; k0 += 4) {
        const int ka = k0 + 2 * half;
        v2f b;
        b.x = breg[(k0 >> 2) * 2 + 0];
        b.y = breg[(k0 >> 2) * 2 + 1];
#pragma unroll
        for (int tile = 0; tile < 2; ++tile) {
            v2f a;
            a.x = Bs[tile * 16 + r][ka];
            a.y = Bs[tile * 16 + r][ka + 1];
            acc[tile] = __builtin_amdgcn_wmma_f32_16x16x4_f32(false, a, false, b,
                                                              (short)0, acc[tile],
                                                              false, false);
        }
    }

    const float bias = bf[col0 + r];
#pragma unroll
    for (int tile = 0; tile < 2; ++tile) {
#pragma unroll
        for (int j = 0; j < 8; ++j) {
            int row = row0 + tile * 16 + j + 8 * half;
            if (row < n_nodes)
                out[(size_t)row * F_DIM + col0 + r] = acc[tile][j] + bias;
        }
    }
}

// ---------------------------------------------------------------------------
// Host-side launcher
// ---------------------------------------------------------------------------
extern "C" void kernel_launch(void* const* d_in, const int* in_sizes, int n_in,
                              void* d_out, int out_size, void* d_ws, size_t ws_size,
                              hipStream_t stream) {
    const float* feat = (const float*)d_in[0];
    const int*   src  = (const int*)d_in[1];
    const int*   dst  = (const int*)d_in[2];
    const float* w1   = (const float*)d_in[3];
    const float* b1   = (const float*)d_in[4];
    const float* w2   = (const float*)d_in[5];
    const float* b2   = (const float*)d_in[6];
    const float* wf   = (const float*)d_in[7];
    const float* bf   = (const float*)d_in[8];

    const int n_nodes = in_sizes[0] / F_DIM;
    const int n_edges = in_sizes[1];

    // Workspace carve-out (256-byte aligned slices)
    char* ws = (char*)d_ws;
    size_t off = 0;
    auto carve = [&](size_t bytes) -> char* {
        char* p = ws + off;
        off = (off + bytes + 255) & ~(size_t)255;
        return p;
    };
    float* proj    = (float*)carve((size_t)2 * n_nodes * F_DIM * sizeof(float));
    float* scores  = (float*)carve((size_t)n_edges * sizeof(float));
    int*   row_ptr = (int*)  carve((size_t)(n_nodes + 1) * sizeof(int));
    float* neigh   = (float*)carve((size_t)n_nodes * F_DIM * sizeof(float));

    const float* proj_dst = proj;
    const float* proj_src = proj + (size_t)n_nodes * F_DIM;

    // K1: dual projection GEMM (WMMA fp32, 64 rows/block)
    {
        int blocks = (n_nodes + 63) / 64;
        proj_kernel<<<dim3(blocks, 2), 256, 0, stream>>>(feat, w1, proj, n_nodes);
    }

    // K2: per-edge scores (1 wave per edge, 8 waves per block)
    {
        int blocks = (n_edges + 7) / 8;
        score_kernel<<<blocks, 256, 0, stream>>>(proj_dst, proj_src, b1, w2, b2,
                                                 src, dst, scores, n_edges);
    }

    // K3: CSR row pointers via binary search on sorted dst
    rowptr_kernel<<<(n_nodes + 1 + 255) / 256, 256, 0, stream>>>(dst, row_ptr,
                                                                 n_nodes, n_edges);

    // K4: segment softmax + weighted aggregation (1 wave per node)
    {
        int blocks = (n_nodes + 7) / 8;
        agg_kernel<<<blocks, 256, 0, stream>>>(feat, scores, src, row_ptr,
                                               neigh, n_nodes);
    }

    // K5: fused output GEMM (K = 256 via two 128-K passes, 32 rows/block) + bias
    {
        int blocks = (n_nodes + 31) / 32;
        out_kernel<<<blocks, 256, 0, stream>>>(feat, neigh, wf, bf,
                                               (float*)d_out, n_nodes);
    }
}